// LSTMModel_12034498363909
// MI455X (gfx1250) — compile-verified
//
#include <hip/hip_runtime.h>

// CDNA5 / gfx1250: wave32, WMMA 16x16 tiles.
typedef float v2f __attribute__((ext_vector_type(2)));
typedef float v4f __attribute__((ext_vector_type(4)));
typedef float v8f __attribute__((ext_vector_type(8)));

#define LDS_STRIDE 36   // floats per row; 36*4B=144B row pitch -> lane*36 mod 64 banks all distinct

// Fast reciprocal: v_rcp_f32 (~1 ulp), avoids the IEEE div_scale/div_fmas/div_fixup chain.
__device__ __forceinline__ float rcp_(float x) {
    return __builtin_amdgcn_rcpf(x);
}
__device__ __forceinline__ float sigmoid_(float x) {
    return rcp_(1.0f + __expf(-x));
}
__device__ __forceinline__ float tanh_(float x) {
    // tanh(x) = 2*sigmoid(2x) - 1 = 2/(1+exp(-2x)) - 1
    return __builtin_fmaf(2.0f, rcp_(1.0f + __expf(-2.0f * x)), -1.0f);
}

__global__ __launch_bounds__(32) void lstm_tail_wmma(
    const float* __restrict__ x,
    const float* __restrict__ Wx0, const float* __restrict__ bx0, const float* __restrict__ bh0,
    const float* __restrict__ Wx1, const float* __restrict__ bx1, const float* __restrict__ bh1,
    const float* __restrict__ Wx2, const float* __restrict__ bx2, const float* __restrict__ bh2,
    const float* __restrict__ Wfc, const float* __restrict__ bfc,
    float* __restrict__ out)
{
    __shared__ float hbuf[16 * LDS_STRIDE];

    const int lane = threadIdx.x;   // 0..31
    const int lc   = lane & 15;     // column within C tile / M row for A
    const int half = lane >> 4;     // 0: lanes 0-15, 1: lanes 16-31
    const int blk  = blockIdx.x;    // 0..63, 16 batch rows each
    const int T = 1024, D = 2;

    // ---------------- layer 0 : gates = x[16,2] @ Wx0^T[2,128] + bx0 + bh0 ----------------
    // A operand (16x4 f32): lanes 0-15 hold K=0,1 (real), lanes 16-31 hold K=2,3 (zero pad)
    v2f a0;
    {
        long row = (long)(blk * 16 + lc);
        long off = (row * (long)T + (T - 1)) * (long)D;   // last timestep only
        float vx0 = x[off + 0];
        float vx1 = x[off + 1];
        a0.x = half ? 0.0f : vx0;
        a0.y = half ? 0.0f : vx1;
    }

    v8f gates[8];
    #pragma unroll
    for (int t = 0; t < 8; ++t) {
        const int n = lc + 16 * t;                 // gate column 0..127
        const float bias = bx0[n] + bh0[n];
        v8f c;
        #pragma unroll
        for (int i = 0; i < 8; ++i) c[i] = bias;   // bias broadcast down column n
        v2f b;
        b.x = Wx0[n * D + half];                   // B v0: K = 0 (lanes 0-15) / 1 (lanes 16-31)
        b.y = 0.0f;                                // B v1: K = 2,3 zero pad
        gates[t] = __builtin_amdgcn_wmma_f32_16x16x4_f32(
            false, a0, false, b, (short)0, c, false, false);
    }

    // cell 0: cx = 0 -> cy = sig(i)*tanh(g); hy = sig(o)*tanh(cy)
    v8f cst[2], h[2];
    #pragma unroll
    for (int s = 0; s < 2; ++s) {
        #pragma unroll
        for (int i = 0; i < 8; ++i) {
            float gi = gates[0 + s][i];
            float gg = gates[4 + s][i];
            float go = gates[6 + s][i];
            float cy = sigmoid_(gi) * tanh_(gg);
            cst[s][i] = cy;
            h[s][i]   = sigmoid_(go) * tanh_(cy);
        }
    }

    // stage h (D layout) into LDS row-major: hbuf[m][col]
    #pragma unroll
    for (int s = 0; s < 2; ++s)
        #pragma unroll
        for (int i = 0; i < 8; ++i)
            hbuf[(i + 8 * half) * LDS_STRIDE + (lc + 16 * s)] = h[s][i];
    __syncthreads();   // single-wave WG: barrier -> S_NOP + enforced ds waits

    // ---------------- layers 1,2 : gates = h[16,32] @ Wx^T[32,128] + bx + bh ----------------
    const float* Wxs[2] = { Wx1, Wx2 };
    const float* bxs[2] = { bx1, bx2 };
    const float* bhs[2] = { bh1, bh2 };

    #pragma unroll
    for (int layer = 0; layer < 2; ++layer) {
        const float* __restrict__ Wx = Wxs[layer];
        const float* __restrict__ bx = bxs[layer];
        const float* __restrict__ bh = bhs[layer];

        #pragma unroll
        for (int t = 0; t < 8; ++t) {
            const int n = lc + 16 * t;
            const float bias = bx[n] + bh[n];
            v8f c;
            #pragma unroll
            for (int i = 0; i < 8; ++i) c[i] = bias;

            // weight row Wx[n, 0..31] as 8 x float4 (16B aligned)
            v4f wrow[8];
            #pragma unroll
            for (int q = 0; q < 8; ++q)
                wrow[q] = *(const v4f*)(Wx + (size_t)n * 32 + q * 4);

            #pragma unroll
            for (int k = 0; k < 8; ++k) {
                // A: h[lc, 4k + 2*half .. +1]  (16x4 f32 layout)
                v2f a = *(const v2f*)(&hbuf[lc * LDS_STRIDE + 4 * k + 2 * half]);
                v2f b;
                b.x = half ? wrow[k][1] : wrow[k][0];   // K = 4k + half
                b.y = half ? wrow[k][3] : wrow[k][2];   // K = 4k + 2 + half
                c = __builtin_amdgcn_wmma_f32_16x16x4_f32(
                    false, a, false, b, (short)0, c, false, false);
            }
            gates[t] = c;
        }

        // cell: cy = c_prev*sig(f) + sig(i)*tanh(g); hy = sig(o)*tanh(cy)
        #pragma unroll
        for (int s = 0; s < 2; ++s) {
            #pragma unroll
            for (int i = 0; i < 8; ++i) {
                float gi = gates[0 + s][i];
                float gf = gates[2 + s][i];
                float gg = gates[4 + s][i];
                float go = gates[6 + s][i];
                float cy = cst[s][i] * sigmoid_(gf) + sigmoid_(gi) * tanh_(gg);
                cst[s][i] = cy;
                h[s][i]   = sigmoid_(go) * tanh_(cy);
            }
        }

        __syncthreads();   // all A reads done before overwriting hbuf
        #pragma unroll
        for (int s = 0; s < 2; ++s)
            #pragma unroll
            for (int i = 0; i < 8; ++i)
                hbuf[(i + 8 * half) * LDS_STRIDE + (lc + 16 * s)] = h[s][i];
        __syncthreads();
    }

    // ---------------- fc : out[m] = h2[m,:] . Wfc + bfc ----------------
    if (lane < 16) {
        float acc = bfc[0];
        #pragma unroll
        for (int j = 0; j < 32; ++j)
            acc += hbuf[lc * LDS_STRIDE + j] * Wfc[j];
        out[blk * 16 + lc] = acc;
    }
}

extern "C" void kernel_launch(void* const* d_in, const int* in_sizes, int n_in,
                              void* d_out, int out_size, void* d_ws, size_t ws_size,
                              hipStream_t stream) {
    (void)in_sizes; (void)n_in; (void)d_ws; (void)ws_size; (void)out_size;
    const float* x   = (const float*)d_in[0];
    const float* Wx0 = (const float*)d_in[1];
    const float* bx0 = (const float*)d_in[2];
    // d_in[3] = Wh0 unused (hx == 0)
    const float* bh0 = (const float*)d_in[4];
    const float* Wx1 = (const float*)d_in[5];
    const float* bx1 = (const float*)d_in[6];
    // d_in[7] = Wh1 unused
    const float* bh1 = (const float*)d_in[8];
    const float* Wx2 = (const float*)d_in[9];
    const float* bx2 = (const float*)d_in[10];
    // d_in[11] = Wh2 unused
    const float* bh2 = (const float*)d_in[12];
    const float* Wfc = (const float*)d_in[13];
    const float* bfc = (const float*)d_in[14];
    float* out = (float*)d_out;

    // B=1024 rows that matter (t = T-1 only), 16 rows per single-wave workgroup
    lstm_tail_wmma<<<64, 32, 0, stream>>>(x, Wx0, bx0, bh0, Wx1, bx1, bh1,
                                          Wx2, bx2, bh2, Wfc, bfc, out);
}